// NMR_25572235281030
// MI455X (gfx1250) — compile-verified
//
#include <hip/hip_runtime.h>
#include <math.h>

// ---------------------------------------------------------------------------
// NMR (PEAQ-style noise-to-mask ratio) forward pass for MI455X / gfx1250.
// Pipeline: init constants -> fused dual real-FFT per frame (LDS radix-2) ->
// band grouping via V_WMMA_F32_16X16X4_F32 (fp32 matrix core, pair-interleaved
// operands for b64 fragment loads) -> freq_spread recursions + deterministic
// reduction -> 10*log10(mean).
// ---------------------------------------------------------------------------

#define THREADS 256

typedef __attribute__((ext_vector_type(2))) float v2f;
typedef __attribute__((ext_vector_type(8))) float v8f;

constexpr int NFFT  = 2048;
constexpr int NBINS = 1025;          // NFFT/2 + 1
constexpr int KPAD  = 1032;          // bins padded: multiple of 8 (clean x2 unroll of K-step 4)
constexpr int PAIRS = KPAD / 2;      // 516 even/odd bin pairs
constexpr int NC    = 109;           // critical bands
constexpr int CPAD  = 112;           // bands padded to multiple of 16 (WMMA M)
constexpr int NB    = 16;            // batch
constexpr int LSIG  = 221184;        // samples per batch row
constexpr int HOPA  = 1024;
constexpr int FPER  = 215;           // frames per batch row
constexpr int FT    = NB * FPER;     // 3440 total frames (multiple of 16, WMMA N)
constexpr float AMAXF = 9.2233720368547758e18f;   // 2^63
constexpr double PI_D = 3.14159265358979323846;

// ---- workspace layout (float offsets, all even -> 8B alignment) -----------
// Ut / X2R / X2E are pair-interleaved: element (k, n) lives at
//   base[ ((k>>1)*N + n)*2 + (k&1) ]   so an (even k, odd k) pair is one b64.
constexpr size_t OFF_WIN  = 0;                               // [NFFT]
constexpr size_t OFF_W2   = OFF_WIN + NFFT;                  // [KPAD]
constexpr size_t OFF_UT   = OFF_W2  + KPAD;                  // [PAIRS][CPAD][2]
constexpr size_t OFF_EIN  = OFF_UT  + (size_t)KPAD * CPAD;   // [CPAD]
constexpr size_t OFF_GM   = OFF_EIN + CPAD;
constexpr size_t OFF_AUC  = OFF_GM  + CPAD;
constexpr size_t OFF_GIL  = OFF_AUC + CPAD;
constexpr size_t OFF_BS   = OFF_GIL + CPAD;
constexpr size_t OFF_X2R  = OFF_BS  + CPAD;                  // [PAIRS][FT][2]
constexpr size_t OFF_X2E  = OFF_X2R + (size_t)KPAD * FT;     // [PAIRS][FT][2]
constexpr size_t OFF_EB   = OFF_X2E + (size_t)KPAD * FT;     // [CPAD][FT]
constexpr size_t OFF_EBN  = OFF_EB  + (size_t)CPAD * FT;     // [CPAD][FT]
constexpr size_t OFF_PART = OFF_EBN + (size_t)CPAD * FT;     // [16] block partials

// ---- band-edge tables (from reference) ------------------------------------
__device__ const float FLc[NC] = {
80.0f,103.445f,127.023f,150.762f,174.694f,198.849f,223.257f,247.95f,272.959f,298.317f,
324.055f,350.207f,376.805f,403.884f,431.478f,459.622f,488.353f,517.707f,547.721f,578.434f,
609.885f,642.114f,675.161f,709.071f,743.884f,779.647f,816.404f,854.203f,893.091f,933.119f,
974.336f,1016.797f,1060.555f,1105.666f,1152.187f,1200.178f,1249.7f,1300.816f,1353.592f,1408.094f,
1464.392f,1522.559f,1582.668f,1644.795f,1709.021f,1775.427f,1844.098f,1915.121f,1988.587f,2064.59f,
2143.227f,2224.597f,2308.806f,2395.959f,2486.169f,2579.551f,2676.223f,2776.309f,2879.937f,2987.238f,
3098.35f,3213.415f,3332.579f,3455.993f,3583.817f,3716.212f,3853.817f,3995.399f,4142.547f,4294.979f,
4452.89f,4616.482f,4785.962f,4961.548f,5143.463f,5331.939f,5527.217f,5729.545f,5939.183f,6156.396f,
6381.463f,6614.671f,6856.316f,7106.708f,7366.166f,7635.02f,7913.614f,8202.302f,8501.454f,8811.45f,
9132.688f,9465.574f,9810.536f,10168.013f,10538.46f,10922.351f,11320.175f,11732.438f,12159.67f,12602.412f,
13061.229f,13536.71f,14029.458f,14540.103f,15069.295f,15617.71f,16186.049f,16775.035f,17385.42f};
__device__ const float FCc[NC] = {
91.708f,115.216f,138.87f,162.702f,186.742f,211.019f,235.566f,260.413f,285.593f,311.136f,
337.077f,363.448f,390.282f,417.614f,445.479f,473.912f,502.95f,532.629f,562.988f,594.065f,
625.899f,658.533f,692.006f,726.362f,761.644f,797.898f,835.17f,873.508f,912.959f,953.576f,
995.408f,1038.511f,1082.938f,1128.746f,1175.995f,1224.744f,1275.055f,1326.992f,1380.623f,1436.014f,
1493.237f,1552.366f,1613.474f,1676.641f,1741.946f,1809.474f,1879.31f,1951.543f,2026.266f,2103.573f,
2183.564f,2266.34f,2352.008f,2440.675f,2532.456f,2627.468f,2725.832f,2827.672f,2933.12f,3042.309f,
3155.379f,3272.475f,3393.745f,3519.344f,3649.432f,3784.176f,3923.748f,4068.324f,4218.09f,4373.237f,
4533.963f,4700.473f,4872.978f,5051.7f,5236.866f,5428.712f,5627.484f,5833.434f,6046.825f,6267.931f,
6497.031f,6734.42f,6980.399f,7235.284f,7499.397f,7773.077f,8056.673f,8350.547f,8655.072f,8970.639f,
9297.648f,9636.52f,9987.683f,10351.586f,10728.695f,11119.49f,11524.47f,11944.149f,12379.066f,12829.775f,
13294.85f,13780.887f,14282.503f,14802.338f,15341.057f,15899.345f,16477.914f,17077.504f,17690.045f};
__device__ const float FUc[NC] = {
103.445f,127.023f,150.762f,174.694f,198.849f,223.257f,247.95f,272.959f,298.317f,324.055f,
350.207f,376.805f,403.884f,431.478f,459.622f,488.353f,517.707f,547.721f,578.434f,609.885f,
642.114f,675.161f,709.071f,743.884f,779.647f,816.404f,854.203f,893.091f,933.113f,974.336f,
1016.797f,1060.555f,1105.666f,1152.187f,1200.178f,1249.7f,1300.816f,1353.592f,1408.094f,1464.392f,
1522.559f,1582.668f,1644.795f,1709.021f,1775.427f,1844.098f,1915.121f,1988.587f,2064.59f,2143.227f,
2224.597f,2308.806f,2395.959f,2486.169f,2579.551f,2676.223f,2776.309f,2879.937f,2987.238f,3098.35f,
3213.415f,3332.579f,3455.993f,3583.817f,3716.212f,3853.348f,3995.399f,4142.547f,4294.979f,4452.89f,
4643.482f,4785.962f,4961.548f,5143.463f,5331.939f,5527.217f,5729.545f,5939.183f,6156.396f,6381.463f,
6614.671f,6856.316f,7106.708f,7366.166f,7635.02f,7913.614f,8202.302f,8501.454f,8811.45f,9132.688f,
9465.574f,9810.536f,10168.013f,10538.46f,10922.351f,11320.175f,11732.438f,12159.67f,12602.412f,13061.229f,
13536.71f,14029.458f,14540.103f,15069.295f,15617.71f,16186.049f,16775.035f,17385.42f,18000.0f};

// ---------------------------------------------------------------------------
// K0: derive all constant tables in double precision (deterministic per call).
// ---------------------------------------------------------------------------
__global__ void nmr_init(float* __restrict__ ws) {
  const int tid = threadIdx.x;

  // GL = 10^(LP/20) / (peak_factor * AMAX/4 * (NFFT-1))
  const double fn  = 1019.5 / 44100.0;
  const double dfn = 1.0 / (double)NFFT;
  const double kk  = floor(fn / dfn);
  const double dfN = fmin((kk + 1.0) * dfn - fn, fn - kk * dfn);
  const double dfW = dfN * (double)(NFFT - 1);
  const double pf  = sin(PI_D * dfW) / (PI_D * dfW * (1.0 - dfW * dfW));
  const double AMAXd = 9223372036854775808.0;
  const double GL  = pow(10.0, 92.0 / 20.0) / (pf * AMAXd / 4.0 * (double)(NFFT - 1));

  for (int i = tid; i < NFFT; i += blockDim.x) {
    double h = 0.5 - 0.5 * cos(2.0 * PI_D * (double)i / (double)(NFFT - 1));
    ws[OFF_WIN + i] = (float)(h * GL);
  }
  for (int k = tid; k < KPAD; k += blockDim.x) {
    float v = 0.0f;
    if (k >= 1 && k < NBINS) {
      double f = ((double)k * (44100.0 / (double)NFFT)) / 1000.0;
      double Adb = -2.184 * pow(f, -0.8) + 6.5 * exp(-0.6 * (f - 3.3) * (f - 3.3))
                   - 0.001 * pow(f, 3.6);
      v = (float)pow(10.0, Adb / 10.0);
    }
    ws[OFF_W2 + k] = v;
  }
  const double dfreq = 44100.0 / (double)NFFT;
  for (size_t idx = tid; idx < (size_t)KPAD * CPAD; idx += blockDim.x) {
    int k = (int)(idx / CPAD), c = (int)(idx % CPAD);
    float v = 0.0f;
    if (c < NC && k < NBINS) {
      double ku = (2.0 * k + 1.0) * 0.5 * dfreq;
      double kl = (2.0 * k - 1.0) * 0.5 * dfreq;
      double ov = fmin((double)FUc[c], ku) - fmax((double)FLc[c], kl);
      v = (float)(fmax(0.0, ov) / dfreq);
    }
    // pair-interleaved U^T: (k, c) -> [((k>>1)*CPAD + c)*2 + (k&1)]
    ws[OFF_UT + ((size_t)(k >> 1) * CPAD + c) * 2 + (k & 1)] = v;
  }
  const double aL = pow(10.0, -2.7 * 0.25);
  for (int c = tid; c < CPAD; c += blockDim.x) {
    if (c < NC) {
      double fc = (double)FCc[c];
      ws[OFF_EIN + c] = (float)pow(10.0, (1.456 * pow(fc / 1000.0, -0.8)) / 10.0);
      double offd = (c <= 48) ? 3.0 : 0.0625 * (double)c;   // 0.25*i*DZ, DZ=0.25
      ws[OFF_GM + c]  = (float)pow(10.0, -offd / 10.0);
      ws[OFF_AUC + c] = (float)pow(10.0, (-2.4 - 23.0 / fc) * 0.25);
      ws[OFF_GIL + c] = (float)((1.0 - pow(aL, (double)(c + 1))) / (1.0 - aL));
    } else {
      ws[OFF_EIN + c] = 0.0f; ws[OFF_GM + c] = 1.0f;
      ws[OFF_AUC + c] = 0.0f; ws[OFF_GIL + c] = 1.0f;
    }
    ws[OFF_BS + c] = 1.0f;
  }
  if (tid == 0) {
    // BS = freq_spread(ones, ones): serial, tiny, double precision
    const double aLe = pow(aL, 0.4);
    double Ene[NC], aUCEe[NC], Esd[NC], r[NC];
    for (int c = 0; c < NC; ++c) {
      double fc  = (double)FCc[c];
      double aUC = pow(10.0, (-2.4 - 23.0 / fc) * 0.25);
      double gIL = (1.0 - pow(aL, (double)(c + 1))) / (1.0 - aL);
      double gIU = (1.0 - pow(aUC, (double)(NC - c))) / (1.0 - aUC);
      Ene[c]   = pow(1.0 / (gIL + gIU - 1.0), 0.4);
      aUCEe[c] = pow(aUC, 0.4);
      r[c] = 0.0;
    }
    double s = 0.0;
    for (int c = NC - 1; c >= 0; --c) { s = aLe * s + Ene[c]; Esd[c] = s; }
    ws[OFF_BS + 0] = (float)pow(Esd[0], 2.5);
    for (int i = 1; i < NC; ++i) {
      r[i - 1] = Ene[i - 1];
      double acc = 0.0;
      for (int m = 0; m < i; ++m) { r[m] *= aUCEe[m]; acc += r[m]; }
      ws[OFF_BS + i] = (float)pow(Esd[i] + acc, 2.5);
    }
  }
}

// ---------------------------------------------------------------------------
// K1: one block per frame. Fused dual real FFT:
//   z = win*round(y*A) + i*win*(round(y*A)-round(x*A))
// 2048-pt radix-2 DIT in 16KB LDS, unpack into ear-weighted power spectra,
// stored pair-interleaved ([PAIRS][FT][2]) with b64 stores as WMMA B operand.
// ---------------------------------------------------------------------------
__global__ __launch_bounds__(THREADS) void nmr_fft(const float* __restrict__ x,
                                                   const float* __restrict__ y,
                                                   float* __restrict__ ws) {
  __shared__ float2 z[NFFT];
  const int tid = threadIdx.x;
  const int ft  = blockIdx.x;
  const int b   = ft / FPER;
  const int t   = ft - b * FPER;
  const size_t base = (size_t)b * LSIG + (size_t)t * HOPA;
  const float* win = ws + OFF_WIN;

  for (int i = tid; i < NFFT; i += THREADS) {
    float sy = y[base + i], sx = x[base + i];
    float ry = rintf(sy * AMAXF);       // identity at these magnitudes (>= 2^23)
    float rx = rintf(sx * AMAXF);
    float w  = win[i];
    int j = (int)(__brev((unsigned)i) >> 21);   // 11-bit reversal
    z[j] = make_float2(ry * w, (ry - rx) * w);
  }
  __syncthreads();

  for (int st = 0; st < 11; ++st) {
    const int hlen = 1 << st;
    for (int q = tid; q < NFFT / 2; q += THREADS) {
      int pos = q & (hlen - 1);
      int grp = q >> st;
      int i0  = (grp << (st + 1)) + pos;
      int i1  = i0 + hlen;
      float ang = -3.14159265358979f * (float)pos / (float)hlen;  // -2*pi*pos/len
      float sw, cw;
      __sincosf(ang, &sw, &cw);
      float2 a  = z[i0];
      float2 bv = z[i1];
      float tr = cw * bv.x - sw * bv.y;
      float ti = cw * bv.y + sw * bv.x;
      z[i1] = make_float2(a.x - tr, a.y - ti);
      z[i0] = make_float2(a.x + tr, a.y + ti);
    }
    __syncthreads();
  }

  float* X2R = ws + OFF_X2R;
  float* X2E = ws + OFF_X2E;
  const float* w2 = ws + OFF_W2;

  auto binPow = [&](int k, float& pr, float& pe) {
    if (k < NBINS) {
      float2 zk = z[k];
      float2 zm = z[(NFFT - k) & (NFFT - 1)];
      float rr = 0.5f * (zk.x + zm.x);          // Xr = (Z[k] + conj(Z[N-k]))/2
      float ri = 0.5f * (zk.y - zm.y);
      float dr = zk.x - zm.x;                   // Xe = (Z[k] - conj(Z[N-k]))/(2i)
      float di = zk.y + zm.y;
      float er = 0.5f * di;
      float ei = -0.5f * dr;
      float wk = w2[k];
      pr = wk * (rr * rr + ri * ri);
      pe = wk * (er * er + ei * ei);
    } else {
      pr = 0.0f; pe = 0.0f;                     // K padding rows
    }
  };

  for (int p = tid; p < PAIRS; p += THREADS) {
    float r0, e0, r1, e1;
    binPow(2 * p,     r0, e0);
    binPow(2 * p + 1, r1, e1);
    const size_t o = ((size_t)p * FT + ft) * 2;
    *(float2*)(X2R + o) = make_float2(r0, r1);  // b64 store
    *(float2*)(X2E + o) = make_float2(e0, e1);  // b64 store
  }
}

// ---------------------------------------------------------------------------
// K2: band grouping as fp32 matrix-core matmul.
//   Eb  = U(112x1032) * X2R(1032x3440),  EbN = U * X2E
// One wave per 16x16 C tile; both accumulators share the A fragment.
// A 16x4 fragment: v0 = K{0|2}, v1 = K{1|3} split at lane 16 (M = lane&15).
// B 4x16 fragment: v0 = row K{0|2}, v1 = row K{1|3} (N = lane&15).
// Pair-interleaved layout makes each fragment a single aligned b64 load.
// C/D: vgpr g holds row g (lanes 0-15) / row g+8 (lanes 16-31).
// ---------------------------------------------------------------------------
__global__ __launch_bounds__(32) void nmr_band_wmma(float* __restrict__ ws) {
  const int lane  = threadIdx.x;
  const int hh    = lane >> 4;
  const int col   = lane & 15;
  const int mbase = blockIdx.x * 16;
  const int nbase = blockIdx.y * 16;
  const float* __restrict__ Ut  = ws + OFF_UT;
  const float* __restrict__ X2R = ws + OFF_X2R;
  const float* __restrict__ X2E = ws + OFF_X2E;

  const size_t aIdx = (size_t)(mbase + col) * 2;
  const size_t bIdx = (size_t)(nbase + col) * 2;

  v8f accR = {};
  v8f accE = {};
#pragma unroll 2
  for (int k0 = 0; k0 < KPAD; k0 += 4) {
    const int kp = (k0 >> 1) + hh;              // this lane's bin-pair index
    v2f a  = *(const v2f*)(Ut  + (size_t)kp * (CPAD * 2) + aIdx);
    v2f br = *(const v2f*)(X2R + (size_t)kp * (FT * 2)   + bIdx);
    v2f be = *(const v2f*)(X2E + (size_t)kp * (FT * 2)   + bIdx);
    __builtin_prefetch(X2R + (size_t)(kp + 2) * (FT * 2) + bIdx, 0, 1);
    accR = __builtin_amdgcn_wmma_f32_16x16x4_f32(false, a, false, br, (short)0,
                                                 accR, false, false);
    accE = __builtin_amdgcn_wmma_f32_16x16x4_f32(false, a, false, be, (short)0,
                                                 accE, false, false);
  }
  float* Eb  = ws + OFF_EB;
  float* EbN = ws + OFF_EBN;
#pragma unroll
  for (int g = 0; g < 8; ++g) {
    const int row = mbase + g + 8 * hh;
    Eb[(size_t)row * FT + nbase + col]  = accR[g];
    EbN[(size_t)row * FT + nbase + col] = accE[g];
  }
}

// ---------------------------------------------------------------------------
// K3: freq_spread per frame (one thread/frame) + EbN/M accumulation,
// deterministic in-block tree reduction -> fixed partial slots.
// ---------------------------------------------------------------------------
__global__ __launch_bounds__(THREADS) void nmr_spread(float* __restrict__ ws) {
  const int tid = threadIdx.x;
  const int ft  = blockIdx.x * THREADS + tid;
  const float* Eb   = ws + OFF_EB;
  const float* EbN  = ws + OFF_EBN;
  const float* EINp = ws + OFF_EIN;
  const float* GMp  = ws + OFF_GM;
  const float* AUCp = ws + OFF_AUC;
  const float* GILp = ws + OFF_GIL;
  const float* BSp  = ws + OFF_BS;

  float localSum = 0.0f;
  if (ft < FT) {
    const float aLe = 0.53703179f;   // (10^(-2.7*DZ))^E_EXP = 10^-0.27
    float Ene[NC], aUCEe[NC], Esd[NC], r[NC];
    for (int c = 0; c < NC; ++c) {
      float E    = fmaxf(Eb[(size_t)c * FT + ft], 1e-12f) + EINp[c];
      float aUCE = AUCp[c] * powf(E, 0.05f);                 // 0.2*DZ
      float gIU  = (1.0f - powf(aUCE, (float)(NC - c))) / (1.0f - aUCE);
      Ene[c]   = powf(E / (GILp[c] + gIU - 1.0f), 0.4f);     // En**e
      aUCEe[c] = powf(aUCE, 0.4f);
    }
    float s = 0.0f;                                          // downward scan
    for (int c = NC - 1; c >= 0; --c) { s = aLe * s + Ene[c]; Esd[c] = s; }
    {
      float Es = powf(Esd[0], 2.5f) / BSp[0];
      localSum += EbN[ft] / (GMp[0] * Es);
    }
    for (int i = 1; i < NC; ++i) {                           // upward running product
      r[i - 1] = Ene[i - 1];
      float acc = 0.0f;
      for (int m = 0; m < i; ++m) { r[m] *= aUCEe[m]; acc += r[m]; }
      float Es = powf(Esd[i] + acc, 2.5f) / BSp[i];
      localSum += EbN[(size_t)i * FT + ft] / (GMp[i] * Es);
    }
  }
  __shared__ float red[THREADS];
  red[tid] = localSum;
  __syncthreads();
  for (int sft = THREADS / 2; sft > 0; sft >>= 1) {
    if (tid < sft) red[tid] += red[tid + sft];
    __syncthreads();
  }
  if (tid == 0) ws[OFF_PART + blockIdx.x] = red[0];
}

__global__ void nmr_finalize(const float* __restrict__ ws, float* __restrict__ out,
                             int nparts) {
  if (threadIdx.x == 0 && blockIdx.x == 0) {
    float s = 0.0f;
    for (int i = 0; i < nparts; ++i) s += ws[OFF_PART + i];
    out[0] = 10.0f * log10f(s / (float)((size_t)FT * NC));
  }
}

// ---------------------------------------------------------------------------
extern "C" void kernel_launch(void* const* d_in, const int* in_sizes, int n_in,
                              void* d_out, int out_size, void* d_ws, size_t ws_size,
                              hipStream_t stream) {
  const float* x = (const float*)d_in[0];
  const float* y = (const float*)d_in[1];
  float* ws  = (float*)d_ws;
  float* out = (float*)d_out;

  nmr_init<<<1, THREADS, 0, stream>>>(ws);
  nmr_fft<<<FT, THREADS, 0, stream>>>(x, y, ws);
  dim3 g2(CPAD / 16, FT / 16);                 // 7 x 215 WMMA tiles
  nmr_band_wmma<<<g2, 32, 0, stream>>>(ws);
  const int nblk = (FT + THREADS - 1) / THREADS;   // 14
  nmr_spread<<<nblk, THREADS, 0, stream>>>(ws);
  nmr_finalize<<<1, 32, 0, stream>>>(ws, out, nblk);
}